// MultiLossFocalLoss_1142461301062
// MI455X (gfx1250) — compile-verified
//
#include <hip/hip_runtime.h>
#include <math.h>

// ---------------------------------------------------------------------------
// Focal-loss reduction for MI455X (gfx1250).
// Memory-bound: 621MB logits + 2x16MB targets @ 23.3 TB/s => ~28us floor.
// Strategy: async global->LDS double-buffered staging (CDNA5 ASYNCcnt path,
// 9x b128 + 1x b32 per thread = exactly 10 async instrs per wave per tile),
// per-lane logsumexp over a conflict-free stride-37 LDS row, deterministic
// two-stage reduction (no float atomics).
// ---------------------------------------------------------------------------

#define NUM_CLASSES   37
#define N_TOTAL       4194304
#define MAIN_THREADS  128
#define TILE_ROWS     128                          // rows per tile
#define TILE_FLOATS   (TILE_ROWS * NUM_CLASSES)    // 4736 floats = 1184 f4
#define TILE_F4       (TILE_FLOATS / 4)            // 1184 = 9*128 + 32
#define F4_ROUNDS     9                            // b128 rounds (uniform)
#define REM_DWORD_OFF (F4_ROUNDS * MAIN_THREADS * 16) // 18432 B: last 128 dwords
#define MAIN_GRID     4096
#define HIST_GRID     1024
#define HIST_THREADS  256

// ln(1/0.99), ln(0.99)
#define LOG_INV_BETA  0.010050335853501441f
#define LOG_BETA      (-0.010050335853501441f)

// ----------------------- CDNA5 async global->LDS ---------------------------
#if defined(__HIP_DEVICE_COMPILE__) && defined(__gfx1250__) && \
    __has_builtin(__builtin_amdgcn_global_load_async_to_lds_b32)
#define HAVE_ASYNC_LDS 1
#else
#define HAVE_ASYNC_LDS 0
#endif

#if HAVE_ASYNC_LDS
typedef int v4i __attribute__((__vector_size__(4 * sizeof(int))));
typedef __attribute__((address_space(1))) int as1_int;
typedef __attribute__((address_space(3))) int as3_int;
typedef __attribute__((address_space(1))) v4i as1_v4i;
typedef __attribute__((address_space(3))) v4i as3_v4i;

#if __has_builtin(__builtin_amdgcn_global_load_async_to_lds_b128)
#define ASYNC_WAIT_N 10  // 9x b128 + 1x b32 per wave per tile

// Template unroll: byte offset must be a frontend-constant immediate.
// The offset applies to BOTH global and LDS addresses (ISA 15.18).
template <int I>
__device__ __forceinline__ void async_copy_b128(as1_v4i* g4, as3_v4i* l4) {
  if constexpr (I < F4_ROUNDS) {
    __builtin_amdgcn_global_load_async_to_lds_b128(g4, l4,
                                                   I * MAIN_THREADS * 16,
                                                   /*cpol=*/0);
    async_copy_b128<I + 1>(g4, l4);
  }
}
#else
#define ASYNC_WAIT_N 37  // fallback: one b32 per dword per thread

template <int I>
__device__ __forceinline__ void async_copy_b32_all(as1_int* g, as3_int* l) {
  if constexpr (I < TILE_FLOATS / MAIN_THREADS) {
    __builtin_amdgcn_global_load_async_to_lds_b32(g, l, I * MAIN_THREADS * 4,
                                                  /*cpol=*/0);
    async_copy_b32_all<I + 1>(g, l);
  }
}
#endif
#endif  // HAVE_ASYNC_LDS

__device__ __forceinline__ void load_tile_async(const float* __restrict__ gsrc,
                                                float* lds_dst, int tid) {
#if HAVE_ASYNC_LDS
#if __has_builtin(__builtin_amdgcn_global_load_async_to_lds_b128)
  // Flat f4 mapping: lane handles f4 index (i*128 + tid); 16B aligned on both
  // sides (tile base 18944*t bytes, lane base tid*16 bytes).
  as1_v4i* g4 = (as1_v4i*)(gsrc) + tid;
  as3_v4i* l4 = (as3_v4i*)(lds_dst) + tid;
  async_copy_b128<0>(g4, l4);
  // Remainder: last 128 dwords, one b32 per lane, uniform across waves.
  as1_int* g1 = (as1_int*)(gsrc) + tid;
  as3_int* l1 = (as3_int*)(lds_dst) + tid;
  __builtin_amdgcn_global_load_async_to_lds_b32(g1, l1, REM_DWORD_OFF,
                                                /*cpol=*/0);
#else
  as1_int* g = (as1_int*)(gsrc) + tid;
  as3_int* l = (as3_int*)(lds_dst) + tid;
  async_copy_b32_all<0>(g, l);
#endif
#else
  // Non-async fallback: synchronous staging (still coalesced).
#pragma unroll
  for (int i = 0; i < TILE_FLOATS / MAIN_THREADS; ++i) {
    const int e = i * MAIN_THREADS + tid;
    lds_dst[e] = gsrc[e];
  }
#endif
}

__device__ __forceinline__ void wait_async_le_tile() {
#if HAVE_ASYNC_LDS
#if __has_builtin(__builtin_amdgcn_s_wait_asynccnt)
  __builtin_amdgcn_s_wait_asynccnt(ASYNC_WAIT_N);
#else
  asm volatile("s_wait_asynccnt %0" ::"i"(ASYNC_WAIT_N) : "memory");
#endif
#endif
}

__device__ __forceinline__ void wait_async_le_0() {
#if HAVE_ASYNC_LDS
#if __has_builtin(__builtin_amdgcn_s_wait_asynccnt)
  __builtin_amdgcn_s_wait_asynccnt(0);
#else
  asm volatile("s_wait_asynccnt 0" ::: "memory");
#endif
#endif
}

// ------------------------------- kernels -----------------------------------

__global__ void __launch_bounds__(64)
init_kernel(unsigned* __restrict__ counts) {
  if (threadIdx.x < NUM_CLASSES) counts[threadIdx.x] = 0u;
}

__global__ void __launch_bounds__(HIST_THREADS)
hist_kernel(const int* __restrict__ target, unsigned* __restrict__ counts) {
  __shared__ unsigned lc[NUM_CLASSES];
  const int tid = threadIdx.x;
  if (tid < NUM_CLASSES) lc[tid] = 0u;
  __syncthreads();
  // Vectorized target reads: N is a multiple of 4*grid*block.
  const int4* t4 = (const int4*)target;
  for (size_t i = (size_t)blockIdx.x * HIST_THREADS + tid; i < (size_t)(N_TOTAL / 4);
       i += (size_t)gridDim.x * HIST_THREADS) {
    const int4 v = t4[i];
    atomicAdd(&lc[v.x], 1u);  // ds_add_u32
    atomicAdd(&lc[v.y], 1u);
    atomicAdd(&lc[v.z], 1u);
    atomicAdd(&lc[v.w], 1u);
  }
  __syncthreads();
  if (tid < NUM_CLASSES) atomicAdd(&counts[tid], lc[tid]);
}

__global__ void __launch_bounds__(MAIN_THREADS)
focal_main(const float* __restrict__ logits, const int* __restrict__ target,
           const unsigned* __restrict__ counts, float* __restrict__ blockSums) {
  __shared__ float tile[2][TILE_FLOATS];
  __shared__ float red[MAIN_THREADS];
  __shared__ unsigned scnt[NUM_CLASSES];

  const int tid = threadIdx.x;
  const int nTiles = N_TOTAL / TILE_ROWS;  // 32768
  const int stride = gridDim.x;
  int t = blockIdx.x;
  int buf = 0;
  float acc = 0.0f;

  // Prologue: start the first tile's async copy, then cache class counts.
  if (t < nTiles) load_tile_async(logits + (size_t)t * TILE_FLOATS, tile[0], tid);
  if (tid < NUM_CLASSES) scnt[tid] = counts[tid];

  for (; t < nTiles; t += stride) {
    const int tn = t + stride;
    if (tn < nTiles) {
      // Issue next tile, then wait until only those remain outstanding:
      // current tile's async loads are complete (they retire in order).
      load_tile_async(logits + (size_t)tn * TILE_FLOATS, tile[buf ^ 1], tid);
      wait_async_le_tile();
    } else {
      wait_async_le_0();
    }
    __syncthreads();  // all waves' portions of tile[buf] are visible

    // One row per lane; stride-37 LDS addressing is bank-conflict free
    // (gcd(37,64)==1).
    const float* row = &tile[buf][tid * NUM_CLASSES];
    float m = -INFINITY;
#pragma unroll
    for (int j = 0; j < NUM_CLASSES; ++j) m = fmaxf(m, row[j]);
    float s = 0.0f;
#pragma unroll
    for (int j = 0; j < NUM_CLASSES; ++j) s += __expf(row[j] - m);

    const int gi = t * TILE_ROWS + tid;
    const int tg = target[gi];
    const float ce = (m + __logf(s)) - row[tg];     // -log_softmax[tgt]
    const float pt = __expf(-ce);
    const float w  = __expf((float)scnt[tg] * LOG_INV_BETA);  // (1/b)^count
    const float omp = 1.0f - pt;
    acc += omp * omp * ce * w;                      // alpha = 1, gamma = 2

    __syncthreads();  // everyone done reading tile[buf] before it is reused
    buf ^= 1;
  }

  // Deterministic block reduction -> one partial per block.
  red[tid] = acc;
  __syncthreads();
#pragma unroll
  for (int w = MAIN_THREADS / 2; w > 0; w >>= 1) {
    if (tid < w) red[tid] += red[tid + w];
    __syncthreads();
  }
  if (tid == 0) blockSums[blockIdx.x] = red[0];
}

__global__ void __launch_bounds__(64)
finalize_kernel(const unsigned* __restrict__ counts,
                const float* __restrict__ blockSums, float* __restrict__ out) {
  __shared__ float red[64];
  const int tid = threadIdx.x;

  float s = 0.0f;
  for (int i = tid; i < MAIN_GRID; i += 64) s += blockSums[i];
  red[tid] = s;
  __syncthreads();
#pragma unroll
  for (int w = 32; w > 0; w >>= 1) {
    if (tid < w) red[tid] += red[tid + w];
    __syncthreads();
  }
  const float focal_mean = red[0] * (1.0f / (float)N_TOTAL);
  __syncthreads();

  // mean over classes of 1/class_weight = beta^(counts/N)
  float c = 0.0f;
  if (tid < NUM_CLASSES)
    c = __expf((float)counts[tid] * (1.0f / (float)N_TOTAL) * LOG_BETA);
  red[tid] = c;
  __syncthreads();
#pragma unroll
  for (int w = 32; w > 0; w >>= 1) {
    if (tid < w) red[tid] += red[tid + w];
    __syncthreads();
  }
  if (tid == 0) out[0] = focal_mean * (red[0] * (1.0f / (float)NUM_CLASSES));
}

// ------------------------------- launcher ----------------------------------

extern "C" void kernel_launch(void* const* d_in, const int* in_sizes, int n_in,
                              void* d_out, int out_size, void* d_ws, size_t ws_size,
                              hipStream_t stream) {
  (void)in_sizes; (void)n_in; (void)out_size; (void)ws_size;
  const float* logits = (const float*)d_in[0];
  const int* target   = (const int*)d_in[1];

  unsigned* counts  = (unsigned*)d_ws;          // 37 (padded to 64) uints
  float* blockSums  = (float*)d_ws + 64;        // MAIN_GRID floats
  float* out        = (float*)d_out;

  init_kernel<<<1, 64, 0, stream>>>(counts);
  hist_kernel<<<HIST_GRID, HIST_THREADS, 0, stream>>>(target, counts);
  focal_main<<<MAIN_GRID, MAIN_THREADS, 0, stream>>>(logits, target, counts, blockSums);
  finalize_kernel<<<1, 64, 0, stream>>>(counts, blockSums, out);
}